// MyGCNEdge_40733469835340
// MI455X (gfx1250) — compile-verified
//
#include <hip/hip_runtime.h>
#include <hip/hip_bf16.h>

#define N_NODES 50000
#define N_EDGES 512000
#define IN_F    7
#define HID     640
#define OUT_F   80

#define KC   160      // K chunk staged in LDS
#define KCP  162      // padded LDS row stride (floats) -> conflict-free ds_load_b64

typedef __attribute__((ext_vector_type(2))) float v2f;
typedef __attribute__((ext_vector_type(8))) float v8f;

static __device__ __forceinline__ void atomic_add_f32(float* p, float v) {
    __hip_atomic_fetch_add(p, v, __ATOMIC_RELAXED, __HIP_MEMORY_SCOPE_AGENT);
}

// ---------------- elementwise / setup kernels ----------------

__global__ void fill_f32(float* __restrict__ p, float v, long long n) {
    long long i = (long long)blockIdx.x * blockDim.x + threadIdx.x;
    if (i < n) p[i] = v;
}

// deg[col[e]] += ew[e]   (deg pre-filled with 1.0 for self-loops)
__global__ void deg_accum(const long long* __restrict__ ei, const float* __restrict__ ew,
                          float* __restrict__ deg) {
    int e = blockIdx.x * blockDim.x + threadIdx.x;
    if (e < N_EDGES) {
        int c = (int)ei[N_EDGES + e];
        atomic_add_f32(&deg[c], ew[e]);
    }
}

__global__ void rsqrt_k(float* __restrict__ deg) {
    int i = blockIdx.x * blockDim.x + threadIdx.x;
    if (i < N_NODES) {
        float d = deg[i];
        deg[i] = d > 0.f ? rsqrtf(d) : 0.f;
    }
}

// nrm[e] = dis[row]*ew*dis[col]
__global__ void norm_k(const long long* __restrict__ ei, const float* __restrict__ ew,
                       const float* __restrict__ dis, float* __restrict__ nrm) {
    int e = blockIdx.x * blockDim.x + threadIdx.x;
    if (e < N_EDGES) {
        int r = (int)ei[e];
        int c = (int)ei[N_EDGES + e];
        nrm[e] = dis[r] * ew[e] * dis[c];
    }
}

// out[n,j] = sum_k x[n,k]*W1[k,j]   (K=7; bias applied post-agg)
__global__ void xw_k(const float* __restrict__ x, const float* __restrict__ W1,
                     float* __restrict__ out) {
    long long idx = (long long)blockIdx.x * blockDim.x + threadIdx.x;
    if (idx >= (long long)N_NODES * HID) return;
    int node = (int)(idx / HID);
    int j    = (int)(idx % HID);
    float acc = 0.f;
#pragma unroll
    for (int k = 0; k < IN_F; ++k)
        acc = fmaf(x[node * IN_F + k], W1[k * HID + j], acc);
    out[idx] = acc;
}

// one block per edge: out[col,f] += nrm[e] * h[row,f]
__global__ void agg_edges(const long long* __restrict__ ei, const float* __restrict__ nrm,
                          const float* __restrict__ h, float* __restrict__ out) {
    int e = blockIdx.x;
    int r = (int)ei[e];
    int c = (int)ei[N_EDGES + e];
    float w = nrm[e];
    const float* src = h + (size_t)r * HID;
    float* dst = out + (size_t)c * HID;
    for (int f = threadIdx.x; f < HID; f += blockDim.x)
        atomic_add_f32(&dst[f], w * src[f]);
}

// out[i,f] = relu(out[i,f] + dis[i]^2 * h[i,f] + b[f])
__global__ void self_bias_relu(float* __restrict__ out, const float* __restrict__ h,
                               const float* __restrict__ dis, const float* __restrict__ b) {
    long long idx = (long long)blockIdx.x * blockDim.x + threadIdx.x;
    if (idx >= (long long)N_NODES * HID) return;
    int i = (int)(idx / HID);
    int f = (int)(idx % HID);
    float d = dis[i];
    float v = out[idx] + d * d * h[idx] + b[f];
    out[idx] = v > 0.f ? v : 0.f;
}

// ---------------- WMMA f32 GEMM with B^T staged in LDS ----------------
// C[M,N] = A[M,K] @ B[K,N], row-major.  Block = 256 thr (8 waves).
// Block tile: 128(M) x 64(N).  Wave tile: 16(M) x 64(N), 4 accumulators.
// LDS holds B^T chunk: bt[n][kk] = B[kc+kk][n0+n], n<64, kk<KC, stride KCP.
// Fragments (wave32, g=lane>>4, lm=lane&15):
//   A 16x4:  v2f = A[m0+lm][k+2g .. k+2g+1]
//   B 4x16:  v2f = { B[k+2g][n], B[k+2g+1][n] }  == bt[n][kk+2g..+1]  (ds_load_b64)
//   C 16x16: component i of v8f = C[m0 + i + 8g][n0 + lm]
__global__ __launch_bounds__(256) void gemm_wmma_lds(const float* __restrict__ A,
                                                     const float* __restrict__ B,
                                                     float* __restrict__ C,
                                                     int M, int N, int K) {
    __shared__ float bt[64 * KCP];
    const int n0    = blockIdx.x * 64;
    const int mbase = blockIdx.y * 128;
    const int wave  = threadIdx.x >> 5;
    const int lane  = threadIdx.x & 31;
    const int m0    = mbase + wave * 16;
    const bool active = (m0 < M);            // wave-uniform
    const int g  = lane >> 4;
    const int lm = lane & 15;

    v8f acc0 = {}, acc1 = {}, acc2 = {}, acc3 = {};
    const float* arow = A + (size_t)(m0 + lm) * K + 2 * g;

    const int ln  = threadIdx.x & 63;        // loader: column within strip
    const int lk0 = threadIdx.x >> 6;        // loader: starting k row (0..3)

    for (int kc = 0; kc < K; kc += KC) {
        __syncthreads();
        // cooperative transpose-load of B chunk (coalesced along n)
        for (int kk = lk0; kk < KC; kk += 4)
            bt[ln * KCP + kk] = B[(size_t)(kc + kk) * N + n0 + ln];
        // prefetch head of next chunk (global_prefetch_b8)
        if (kc + KC < K)
            __builtin_prefetch(&B[(size_t)(kc + KC + lk0) * N + n0 + ln], 0, 1);
        __syncthreads();

        if (active) {
            for (int kk = 0; kk < KC; kk += 4) {
                v2f a = *(const v2f*)(arow + kc + kk);
                const int kb = kk + 2 * g;
                v2f b0 = *(const v2f*)&bt[(lm +  0) * KCP + kb];
                v2f b1 = *(const v2f*)&bt[(lm + 16) * KCP + kb];
                v2f b2 = *(const v2f*)&bt[(lm + 32) * KCP + kb];
                v2f b3 = *(const v2f*)&bt[(lm + 48) * KCP + kb];
                acc0 = __builtin_amdgcn_wmma_f32_16x16x4_f32(false, a, false, b0, (short)0, acc0, false, false);
                acc1 = __builtin_amdgcn_wmma_f32_16x16x4_f32(false, a, false, b1, (short)0, acc1, false, false);
                acc2 = __builtin_amdgcn_wmma_f32_16x16x4_f32(false, a, false, b2, (short)0, acc2, false, false);
                acc3 = __builtin_amdgcn_wmma_f32_16x16x4_f32(false, a, false, b3, (short)0, acc3, false, false);
            }
        }
    }

    if (active) {
#pragma unroll
        for (int i = 0; i < 8; ++i) {
            float* cr = C + (size_t)(m0 + i + 8 * g) * N + n0 + lm;
            cr[0]  = acc0[i];
            cr[16] = acc1[i];
            cr[32] = acc2[i];
            cr[48] = acc3[i];
        }
    }
}

// out[m, 0:80] = concat(h2, identity)[m, :] @ We + be   (virtual K = 1280)
// Per 160-chunk: A base switches (uniformly) between H2 and ID; matching
// 160x80 We slice is transposed into LDS.  Wave tile: 16(M) x 80(N), 5 acc.
__global__ __launch_bounds__(256) void embed_wmma_lds(const float* __restrict__ H2,
                                                      const float* __restrict__ ID,
                                                      const float* __restrict__ We,
                                                      const float* __restrict__ be,
                                                      float* __restrict__ out) {
    __shared__ float bt[80 * KCP];
    const int mbase = blockIdx.x * 128;
    const int wave  = threadIdx.x >> 5;
    const int lane  = threadIdx.x & 31;
    const int m0    = mbase + wave * 16;
    const bool active = (m0 < N_NODES);      // wave-uniform
    const int g  = lane >> 4;
    const int lm = lane & 15;

    v8f acc[5];
    {
        // init with bias: component i = row, column = t*16+lm
        float bv0 = be[ 0 + lm], bv1 = be[16 + lm], bv2 = be[32 + lm],
              bv3 = be[48 + lm], bv4 = be[64 + lm];
#pragma unroll
        for (int i = 0; i < 8; ++i) {
            acc[0][i] = bv0; acc[1][i] = bv1; acc[2][i] = bv2;
            acc[3][i] = bv3; acc[4][i] = bv4;
        }
    }

    for (int chunk = 0; chunk < (2 * HID) / KC; ++chunk) {   // 8 chunks
        const int kv0 = chunk * KC;                          // virtual K offset
        const float* abase = (kv0 < HID)
            ? H2 + (size_t)(m0 + lm) * HID + kv0 + 2 * g
            : ID + (size_t)(m0 + lm) * HID + (kv0 - HID) + 2 * g;

        __syncthreads();
        for (int idx = threadIdx.x; idx < 80 * KC; idx += 256) {
            int kk = idx / 80;
            int n  = idx - kk * 80;
            bt[n * KCP + kk] = We[(size_t)(kv0 + kk) * OUT_F + n];
        }
        __syncthreads();

        if (active) {
            for (int kk = 0; kk < KC; kk += 4) {
                v2f a = *(const v2f*)(abase + kk);
                const int kb = kk + 2 * g;
#pragma unroll
                for (int t = 0; t < 5; ++t) {
                    v2f b = *(const v2f*)&bt[(lm + 16 * t) * KCP + kb];
                    acc[t] = __builtin_amdgcn_wmma_f32_16x16x4_f32(false, a, false, b, (short)0, acc[t], false, false);
                }
            }
        }
    }

    if (active) {
#pragma unroll
        for (int i = 0; i < 8; ++i) {
            float* cr = out + (size_t)(m0 + i + 8 * g) * OUT_F + lm;
#pragma unroll
            for (int t = 0; t < 5; ++t) cr[t * 16] = acc[t][i];
        }
    }
}

// ---------------- launch ----------------

extern "C" void kernel_launch(void* const* d_in, const int* in_sizes, int n_in,
                              void* d_out, int out_size, void* d_ws, size_t ws_size,
                              hipStream_t stream) {
    const float*     x  = (const float*)d_in[0];
    const long long* ei = (const long long*)d_in[1];   // [2, E] int64: row=ei[e], col=ei[E+e]
    const float*     ew = (const float*)d_in[2];
    const float*     W1 = (const float*)d_in[3];
    const float*     b1 = (const float*)d_in[4];
    const float*     W2 = (const float*)d_in[5];
    const float*     b2 = (const float*)d_in[6];
    const float*     We = (const float*)d_in[7];
    const float*     be = (const float*)d_in[8];
    float* out = (float*)d_out;

    float* ws   = (float*)d_ws;
    float* deg  = ws;                                   // N  (becomes dis)
    float* nrm  = deg + N_NODES;                        // E
    float* bufA = nrm + N_EDGES;                        // N*HID : xw, then g = identity@W2
    float* bufB = bufA + (size_t)N_NODES * HID;         // N*HID : identity
    float* bufC = bufB + (size_t)N_NODES * HID;         // N*HID : h2

    const long long NH = (long long)N_NODES * HID;
    const int TB = 256;
    const int gN  = (N_NODES + TB - 1) / TB;
    const int gE  = (N_EDGES + TB - 1) / TB;
    const int gNH = (int)((NH + TB - 1) / TB);
    const int mblocks = (N_NODES + 127) / 128;          // 391

    // normalization coefficients
    fill_f32<<<gN, TB, 0, stream>>>(deg, 1.0f, N_NODES);
    deg_accum<<<gE, TB, 0, stream>>>(ei, ew, deg);
    rsqrt_k<<<gN, TB, 0, stream>>>(deg);                // deg -> dis
    norm_k<<<gE, TB, 0, stream>>>(ei, ew, deg, nrm);

    // conv1: xw = x@W1; identity = relu(agg(xw) + b1)
    xw_k<<<gNH, TB, 0, stream>>>(x, W1, bufA);
    fill_f32<<<gNH, TB, 0, stream>>>(bufB, 0.0f, NH);
    agg_edges<<<N_EDGES, TB, 0, stream>>>(ei, nrm, bufA, bufB);
    self_bias_relu<<<gNH, TB, 0, stream>>>(bufB, bufA, deg, b1);   // bufB = identity

    // conv2: g = identity@W2 (WMMA); h2 = relu(agg(g) + b2)
    {
        dim3 grid(HID / 64, mblocks);                   // (10, 391)
        gemm_wmma_lds<<<grid, TB, 0, stream>>>(bufB, W2, bufA, N_NODES, HID, HID);
    }
    fill_f32<<<gNH, TB, 0, stream>>>(bufC, 0.0f, NH);
    agg_edges<<<N_EDGES, TB, 0, stream>>>(ei, nrm, bufA, bufC);
    self_bias_relu<<<gNH, TB, 0, stream>>>(bufC, bufA, deg, b2);   // bufC = h2

    // embed: out = [h2, identity] @ We + be (WMMA, LDS-staged We)
    embed_wmma_lds<<<mblocks, TB, 0, stream>>>(bufC, bufB, We, be, out);
}